// SAGEClassifier_89781996356214
// MI455X (gfx1250) — compile-verified
//
#include <hip/hip_runtime.h>
#include <hip/hip_bf16.h>

typedef __attribute__((ext_vector_type(2))) float v2f;
typedef __attribute__((ext_vector_type(8))) float v8f;

#define NNODES 50000   // multiple of 16 -> GEMM row tiles are always full
#define NEDGES 800000
#define DIM    128

static_assert(NNODES % 16 == 0, "row tiles must be exact for unguarded WMMA stores");

// ---------------------------------------------------------------- utilities
__global__ void zero_f32(float* __restrict__ p, long long n) {
  long long i = (long long)blockIdx.x * blockDim.x + threadIdx.x;
  long long stride = (long long)gridDim.x * blockDim.x;
  for (; i < n; i += stride) p[i] = 0.0f;
}

__global__ void deg_count(const int* __restrict__ dst, float* __restrict__ deg, int n_edges) {
  int i = blockIdx.x * blockDim.x + threadIdx.x;
  if (i < n_edges) atomicAdd(&deg[dst[i]], 1.0f);
}

__global__ void deg_inv(float* __restrict__ deg, int n) {
  int i = blockIdx.x * blockDim.x + threadIdx.x;
  if (i < n) deg[i] = 1.0f / fmaxf(deg[i], 1.0f);
}

// one thread per (edge, feature): coalesced read of feat[src], coalesced
// global_atomic_add_f32 into agg[dst]
__global__ void scatter_add(const float* __restrict__ feat, const int* __restrict__ src,
                            const int* __restrict__ dst, float* __restrict__ agg,
                            long long total) {
  long long i = (long long)blockIdx.x * blockDim.x + threadIdx.x;
  if (i >= total) return;
  int e = (int)(i >> 7);       // DIM == 128
  int f = (int)(i & 127);
  int s = src[e];
  int d = dst[e];
  atomicAdd(&agg[(long long)d * DIM + f], feat[(long long)s * DIM + f]);
}

// ------------------------------------------------- WMMA fused SAGE linear
// out = relu( (agg * inv_deg) @ Wl + bias + x @ Wr ), all [N,128] fp32.
// One wave -> one 16x16 output tile; two V_WMMA_F32_16X16X4_F32 chains per
// K step, K = 128 (32 steps). All control flow is wave-uniform (EXEC all 1s),
// and all tiles are full (NNODES % 16 == 0) so loads/stores are unguarded.
__global__ void __launch_bounds__(256)
sage_linear_wmma(const float* __restrict__ agg, const float* __restrict__ inv_deg,
                 const float* __restrict__ xin,
                 const float* __restrict__ Wl, const float* __restrict__ bias,
                 const float* __restrict__ Wr, float* __restrict__ out, int n_nodes) {
  const int lane = threadIdx.x & 31;
  const int wave = (blockIdx.x * blockDim.x + threadIdx.x) >> 5;
  const int row_tile = wave >> 3;          // 8 column tiles of 16 across DIM=128
  const int col_tile = wave & 7;
  const int m0 = row_tile * 16;
  if (m0 >= n_nodes) return;               // wave-uniform exit

  const int half = lane >> 4;              // 0: K+{0,1}, 1: K+{2,3}
  const int mr   = lane & 15;
  const int m = m0 + mr;                   // always < n_nodes (full tiles)
  const int n = col_tile * 16 + mr;

  const float scale = inv_deg[m];
  const float* __restrict__ aggRow = agg + (long long)m * DIM;
  const float* __restrict__ xRow   = xin + (long long)m * DIM;

  v8f acc = {};
  #pragma unroll 4
  for (int k = 0; k < DIM; k += 4) {
    const int ka = k + half * 2;
    v2f a0, b0, a1, b1v;
    // A fragment: two contiguous K values -> single b64 load
    a0[0] = aggRow[ka] * scale;
    a0[1] = aggRow[ka + 1] * scale;
    b0[0] = Wl[(ka + 0) * DIM + n];
    b0[1] = Wl[(ka + 1) * DIM + n];
    acc = __builtin_amdgcn_wmma_f32_16x16x4_f32(false, a0, false, b0,
                                                (short)0, acc, false, false);
    a1[0] = xRow[ka];
    a1[1] = xRow[ka + 1];
    b1v[0] = Wr[(ka + 0) * DIM + n];
    b1v[1] = Wr[(ka + 1) * DIM + n];
    acc = __builtin_amdgcn_wmma_f32_16x16x4_f32(false, a1, false, b1v,
                                                (short)0, acc, false, false);
  }

  // D layout: lane l, VGPR v -> row m0 + v + (l/16)*8, col n. All rows valid;
  // straight-line coalesced b32 stores (16 lanes cover 16 consecutive cols).
  const float bn = bias[n];
  const long long base = (long long)(m0 + half * 8) * DIM + n;
  #pragma unroll
  for (int v = 0; v < 8; ++v) {
    const float val = acc[v] + bn;
    out[base + (long long)v * DIM] = val > 0.0f ? val : 0.0f;
  }
}

// ------------------------------------------------- edge classifier head
// One wave32 per edge: coalesced gather of h[src], h[dst] (4 floats/lane),
// dual dot products vs W_fc columns, shfl reduction, 2-way log_softmax.
__global__ void __launch_bounds__(256)
edge_score(const float* __restrict__ h, const int* __restrict__ src,
           const int* __restrict__ dst, const float* __restrict__ Wfc,
           const float* __restrict__ bfc, float* __restrict__ out, int n_edges) {
  const int wave = (blockIdx.x * blockDim.x + threadIdx.x) >> 5;
  const int lane = threadIdx.x & 31;
  if (wave >= n_edges) return;
  const int s = src[wave];
  const int d = dst[wave];
  const float* __restrict__ hs = h + (long long)s * DIM;
  const float* __restrict__ hd = h + (long long)d * DIM;

  float acc0 = 0.0f, acc1 = 0.0f;
  #pragma unroll
  for (int i = lane; i < DIM; i += 32) {
    const float vs = hs[i];
    const float vd = hd[i];
    acc0 += vs * Wfc[i * 2 + 0] + vd * Wfc[(DIM + i) * 2 + 0];
    acc1 += vs * Wfc[i * 2 + 1] + vd * Wfc[(DIM + i) * 2 + 1];
  }
  #pragma unroll
  for (int off = 16; off > 0; off >>= 1) {
    acc0 += __shfl_down(acc0, off, 32);
    acc1 += __shfl_down(acc1, off, 32);
  }
  if (lane == 0) {
    const float s0 = acc0 + bfc[0];
    const float s1 = acc1 + bfc[1];
    const float mx = fmaxf(s0, s1);
    const float lse = mx + logf(expf(s0 - mx) + expf(s1 - mx));
    out[(long long)wave * 2 + 0] = s0 - lse;
    out[(long long)wave * 2 + 1] = s1 - lse;
  }
}

// ---------------------------------------------------------------- launcher
extern "C" void kernel_launch(void* const* d_in, const int* in_sizes, int n_in,
                              void* d_out, int out_size, void* d_ws, size_t ws_size,
                              hipStream_t stream) {
  const float* x   = (const float*)d_in[0];
  const int*   ei  = (const int*)d_in[1];
  const float* W1l = (const float*)d_in[2];
  const float* b1  = (const float*)d_in[3];
  const float* W1r = (const float*)d_in[4];
  const float* W2l = (const float*)d_in[5];
  const float* b2  = (const float*)d_in[6];
  const float* W2r = (const float*)d_in[7];
  const float* Wfc = (const float*)d_in[8];
  const float* bfc = (const float*)d_in[9];
  float* out = (float*)d_out;

  const int* src = ei;            // edge_index[0]
  const int* dst = ei + NEDGES;   // edge_index[1]

  char* ws = (char*)d_ws;
  size_t off = 0;
  auto salloc = [&](size_t bytes) -> char* {
    char* p = ws + off;
    off += (bytes + 255) & ~(size_t)255;
    return p;
  };
  float* deg = (float*)salloc((size_t)NNODES * sizeof(float));
  float* agg = (float*)salloc((size_t)NNODES * DIM * sizeof(float));
  float* h1  = (float*)salloc((size_t)NNODES * DIM * sizeof(float));
  float* h2  = (float*)salloc((size_t)NNODES * DIM * sizeof(float));
  (void)ws_size; (void)in_sizes; (void)n_in; (void)out_size;

  const long long nfeat = (long long)NNODES * DIM;       // 6.4M
  const long long etot  = (long long)NEDGES * DIM;       // 102.4M
  const int tiles = NNODES / 16;                         // 3125
  const int gemm_blocks = (tiles * 8 + 7) / 8;           // 8 waves/block

  // degrees (shared by both layers)
  zero_f32<<<256, 256, 0, stream>>>(deg, NNODES);
  deg_count<<<(NEDGES + 255) / 256, 256, 0, stream>>>(dst, deg, NEDGES);
  deg_inv<<<(NNODES + 255) / 256, 256, 0, stream>>>(deg, NNODES);

  // layer 1
  zero_f32<<<2048, 256, 0, stream>>>(agg, nfeat);
  scatter_add<<<(unsigned)((etot + 255) / 256), 256, 0, stream>>>(x, src, dst, agg, etot);
  sage_linear_wmma<<<gemm_blocks, 256, 0, stream>>>(agg, deg, x, W1l, b1, W1r, h1, NNODES);

  // layer 2
  zero_f32<<<2048, 256, 0, stream>>>(agg, nfeat);
  scatter_add<<<(unsigned)((etot + 255) / 256), 256, 0, stream>>>(h1, src, dst, agg, etot);
  sage_linear_wmma<<<gemm_blocks, 256, 0, stream>>>(agg, deg, h1, W2l, b2, W2r, h2, NNODES);

  // edge scoring (one wave per edge)
  edge_score<<<(NEDGES + 7) / 8, 256, 0, stream>>>(h2, src, dst, Wfc, bfc, out, NEDGES);
}